// ForceAggregation_48747878810162
// MI455X (gfx1250) — compile-verified
//
#include <hip/hip_runtime.h>
#include <stdint.h>

// ---------------------------------------------------------------------------
// Batched 96x96 fp32 Hessian mat-vec + damping for MI455X (gfx1250).
//
//   f[mol] = H[mol] @ x[mol] + DAMP_FACTOR * x[mol],   mol = 0..8191, D = 96
//
// HBM-bound (0.5 FLOP/byte, 302 MB of Hessian read once -> ~13 us floor at
// 23.3 TB/s). Data path: TDM (tensor_load_to_lds) DMAs each 36 KB Hessian
// block into LDS with hardware padding (+2 dwords per 32 dwords -> LDS row
// stride 102 dwords, bank-conflict-free for the strided A-fragment reads),
// then 6 wave32s per workgroup run chained V_WMMA_F32_16X16X4_F32 ops.
// B is the x-chunk replicated across all 16 N-columns and C is initialized
// to DAMP*x, so every column of D equals y and damping costs nothing.
// ---------------------------------------------------------------------------

typedef float v2f __attribute__((ext_vector_type(2)));
typedef float v8f __attribute__((ext_vector_type(8)));
typedef unsigned int v4u __attribute__((ext_vector_type(4)));
typedef int v4i __attribute__((ext_vector_type(4)));
typedef int v8i __attribute__((ext_vector_type(8)));

#define DQ        96     // Hessian block dim = 3 * 32 atoms
#define ROW_DW    102    // padded LDS row stride in dwords (96 + 3 pads * 2)
#define HTILE_DW  (ROW_DW * DQ)   // 9792 dwords
// TDM pad: every 32 dwords stored, skip 2 dwords. Element (r,k) lands at
// dword 102*r + k + 2*(k>>5). Stride 102 dwords == 38 banks (mod 64):
// 19*delta % 32 != 0 for delta in 1..15 -> 16-lane reads are conflict-free.

#define DAMP_FACTOR ((float)(0.1 * 627.5094740631 / (0.529177210903 * 0.529177210903)))

__global__ __launch_bounds__(192) void ForceAggregation_48747878810162_kernel(
    const float* __restrict__ ns,     // [M*32, 3] == [M, 96] flat
    const float* __restrict__ hess,   // [M*96, 96] row-major
    float* __restrict__ out)          // [M*32, 3] == [M, 96] flat
{
    __shared__ __align__(16) float sh_h[HTILE_DW + 4];  // +4: trailing TDM pad slack
    __shared__ __align__(16) float sh_x[DQ];

    const int mol  = blockIdx.x;
    const int tid  = threadIdx.x;
    const int wave = tid >> 5;        // 0..5 : which 16-row tile of H
    const int lane = tid & 31;
    const size_t hbase = (size_t)mol * (DQ * DQ);

    // Stage x (96 contiguous floats) into LDS.
    if (tid < DQ) sh_x[tid] = ns[(size_t)mol * DQ + tid];

#if __has_builtin(__builtin_amdgcn_tensor_load_to_lds)
    // --- TDM: DMA the whole 96x96 fp32 block into LDS with padding --------
    if (wave == 0) {
        const uint64_t gaddr = (uint64_t)(uintptr_t)(hess + hbase);
        const uint32_t laddr = (uint32_t)(uintptr_t)(&sh_h[0]); // low 32b = LDS offset

        v4u g0;
        g0[0] = 1u;                                   // count=1 (valid descriptor)
        g0[1] = laddr;                                // lds_addr
        g0[2] = (uint32_t)gaddr;                      // global_addr[31:0]
        g0[3] = (uint32_t)((gaddr >> 32) & 0x01FFFFFFu) // global_addr[56:32]
              | 0x80000000u;                          // type = 2 ("image")

        v8i g1;
        g1[0] = (int)((2u << 16)      // data_size = 4 bytes
                    | (1u << 20)      // pad_enable
                    | (4u << 22)      // pad_interval: every 32 dwords
                    | (1u << 25));    // pad_amount: 2 dwords
        g1[1] = (int)(96u << 16);     // tensor_dim0 = 96 (bits 79:48)
        g1[2] = (int)(96u << 16);     // tensor_dim1 = 96 (bits 111:80)
        g1[3] = (int)(96u << 16);     // tile_dim0   = 96 (bits 127:112)
        g1[4] = 96;                   // tile_dim1   = 96 (bits 143:128)
        g1[5] = 96;                   // tensor_dim0_stride = 96 elements
        g1[6] = 0;                    // stride hi / tensor_dim1_stride lo
        g1[7] = 0;

        v4i g2 = {0, 0, 0, 0};        // 2D tensor: groups 2/3 unused
        v4i g3 = {0, 0, 0, 0};
        v8i g4 = {0, 0, 0, 0, 0, 0, 0, 0};  // extra group (unused, 6-arg form)

        __builtin_amdgcn_tensor_load_to_lds(g0, g1, g2, g3, g4, 0);
        __builtin_amdgcn_s_wait_tensorcnt(0);
    }
#else
    // Fallback: coalesced cooperative copy into the same padded layout.
    for (int idx = tid; idx < DQ * DQ; idx += 192) {
        const int r = idx / DQ;
        const int k = idx - r * DQ;
        sh_h[ROW_DW * r + k + 2 * (k >> 5)] = hess[hbase + idx];
    }
#endif

    __syncthreads();

    // --- WMMA phase: wave t computes y[16t .. 16t+15] ----------------------
    // A frag (16x4 f32): lanes 0-15 hold M=lane, K=k0/k0+1; lanes 16-31 hold
    // M=lane-16, K=k0+2/k0+3  ->  contiguous float2 per lane.
    const int m    = lane & 15;
    const int h    = lane >> 4;               // 0 or 1 (lane half)
    const int row  = wave * 16 + m;           // H row this lane feeds
    const int cbas = wave * 16 + h * 8;       // C/D: VGPR v holds M = cbas+v

    // Fold damping into the accumulator: every column of C = DAMP * x-chunk.
    v8f acc;
#pragma unroll
    for (int v = 0; v < 8; ++v) acc[v] = DAMP_FACTOR * sh_x[cbas + v];

#pragma unroll
    for (int kk = 0; kk < 24; ++kk) {
        const int kcol = kk * 4 + 2 * h;      // even -> 8B-aligned LDS reads
        v2f a = *(const v2f*)(&sh_h[ROW_DW * row + kcol + 2 * (kcol >> 5)]);
        v2f b = *(const v2f*)(&sh_x[kcol]);   // x replicated across all N cols
        acc = __builtin_amdgcn_wmma_f32_16x16x4_f32(
            /*neg_a=*/false, a, /*neg_b=*/false, b,
            /*c_mod=*/(short)0, acc, /*reuse_a=*/false, /*reuse_b=*/false);
    }

    // Every column of D equals y; lanes with N==0 (lane 0 and lane 16) store.
    if ((lane & 15) == 0) {
        float* p = out + (size_t)mol * DQ + wave * 16 + h * 8;
#pragma unroll
        for (int v = 0; v < 8; ++v) p[v] = acc[v];
    }
}

extern "C" void kernel_launch(void* const* d_in, const int* in_sizes, int n_in,
                              void* d_out, int out_size, void* d_ws, size_t ws_size,
                              hipStream_t stream) {
    const float* ns   = (const float*)d_in[0];   // [M*32, 3] fp32
    const float* hess = (const float*)d_in[1];   // [M*96, 96] fp32
    // d_in[2] (idx_m) and d_in[3] (n_atoms) are uniform -> unused on device.
    float* out = (float*)d_out;                  // [M*32, 3] fp32

    const int M = (n_in >= 4) ? in_sizes[3] : 8192;   // number of molecules

    ForceAggregation_48747878810162_kernel<<<M, 192, 0, stream>>>(ns, hess, out);
    (void)d_ws; (void)ws_size; (void)out_size;
}